// LargeKernelAttention_27668179321445
// MI455X (gfx1250) — compile-verified
//
#include <hip/hip_runtime.h>
#include <hip/hip_bf16.h>

#define NB   32
#define CCH  384
#define HH   56
#define WWX  56
#define HWP  (56 * 56)   // 3136
#define KS   21
#define PADK 10

typedef float v2f  __attribute__((ext_vector_type(2)));
typedef float v8f  __attribute__((ext_vector_type(8)));
typedef unsigned int u32x4 __attribute__((ext_vector_type(4)));
typedef int   i32x4 __attribute__((ext_vector_type(4)));
typedef int   i32x8 __attribute__((ext_vector_type(8)));
typedef int   gv4i  __attribute__((vector_size(16)));   // matches builtin's V4i

// ---- CDNA5 async/TDM feature detection (fallback keeps it compiling) ------
#if defined(__has_builtin)
#  if __has_builtin(__builtin_amdgcn_global_load_async_to_lds_b128)
#    define HAVE_ASYNC_LDS 1
#  endif
#  if __has_builtin(__builtin_amdgcn_tensor_load_to_lds)
#    define HAVE_TDM 1
#  endif
#endif
#ifndef HAVE_ASYNC_LDS
#  define HAVE_ASYNC_LDS 0
#endif
#ifndef HAVE_TDM
#  define HAVE_TDM 0
#endif

#if defined(__has_builtin) && __has_builtin(__builtin_amdgcn_s_wait_asynccnt)
#  define WAIT_ASYNC0() __builtin_amdgcn_s_wait_asynccnt(0)
#else
#  define WAIT_ASYNC0() asm volatile("s_wait_asynccnt 0x0" ::: "memory")
#endif
#if defined(__has_builtin) && __has_builtin(__builtin_amdgcn_s_wait_tensorcnt)
#  define WAIT_TENSOR0() __builtin_amdgcn_s_wait_tensorcnt(0)
#else
#  define WAIT_TENSOR0() asm volatile("s_wait_tensorcnt 0x0" ::: "memory")
#endif

// async-LDS builtin wants (AS1 v4i*, AS3 v4i*, imm, imm)
#define GAS1_V4(p) ((__attribute__((address_space(1))) gv4i*)(p))
#define LAS3_V4(p) ((__attribute__((address_space(3))) gv4i*)(p))
#define AS3(p)     ((__attribute__((address_space(3))) void*)(p))

// ---------------------------------------------------------------------------
// Kernel 1: depthwise 21x21 conv (groups=C) + bias  ->  attn (workspace)
// grid: (N*C, 2 strips of 28 rows), block 224 threads (7 waves).
// Each active thread: 2 rows x 4 cols -> 8 independent FMA chains per tap
// (VOPD v_dual_fmac fodder); 168 FMA per 18 b128 LDS loads per tap-row.
// ---------------------------------------------------------------------------
__global__ __launch_bounds__(224) void dw_conv_kernel(
    const float* __restrict__ x, const float* __restrict__ dw_w,
    const float* __restrict__ dw_b, float* __restrict__ attn)
{
  __shared__ float tile[48 * 80];   // (28+20) rows x (56+20) cols, stride 80
  __shared__ float wlds[21 * 24];   // padded weight rows (16B aligned)

  const int bc    = blockIdx.x;           // n*C + c
  const int c     = bc % CCH;
  const int h0    = blockIdx.y * 28;
  const int tid   = threadIdx.x;
  const int plane = bc * HWP;

  // stage input tile with zero halo
  for (int e = tid; e < 48 * 76; e += 224) {
    int r  = e / 76;
    int cc = e - r * 76;
    int gh = h0 + r - PADK;
    int gw = cc - PADK;
    float v = 0.0f;
    if ((unsigned)gh < (unsigned)HH && (unsigned)gw < (unsigned)WWX)
      v = x[plane + gh * WWX + gw];
    tile[r * 80 + cc] = v;
  }
  // stage weights (row-padded to 24 for aligned b128 reads)
  for (int e = tid; e < KS * KS; e += 224) {
    int kh = e / KS;
    int kw = e - kh * KS;
    wlds[kh * 24 + kw] = dw_w[c * (KS * KS) + e];
  }
  __syncthreads();

  if (tid < 196) {
    const int wq = tid % 14;          // 4-col group
    const int hq = tid / 14;          // 2-row group (0..13)
    float accA[4] = {0.f, 0.f, 0.f, 0.f};
    float accB[4] = {0.f, 0.f, 0.f, 0.f};

    for (int kh = 0; kh < KS; ++kh) {
      const int r0 = 2 * hq + kh;
      const float* trow0 = &tile[(r0 + 0) * 80 + wq * 4];
      const float* trow1 = &tile[(r0 + 1) * 80 + wq * 4];

      float win0[24], win1[24], wt[24];
      float4* a4 = reinterpret_cast<float4*>(win0);
      float4* b4 = reinterpret_cast<float4*>(win1);
      float4* t4 = reinterpret_cast<float4*>(wt);
      const float* wrow = &wlds[kh * 24];
      #pragma unroll
      for (int i = 0; i < 6; ++i) {
        a4[i] = *reinterpret_cast<const float4*>(trow0 + 4 * i);
        b4[i] = *reinterpret_cast<const float4*>(trow1 + 4 * i);
        t4[i] = *reinterpret_cast<const float4*>(wrow + 4 * i);
      }
      #pragma unroll
      for (int kw = 0; kw < KS; ++kw) {
        #pragma unroll
        for (int j = 0; j < 4; ++j) {
          accA[j] = fmaf(win0[kw + j], wt[kw], accA[j]);
          accB[j] = fmaf(win1[kw + j], wt[kw], accB[j]);
        }
      }
    }
    const float b = dw_b[c];
    float4 oa, ob;
    oa.x = accA[0] + b; oa.y = accA[1] + b; oa.z = accA[2] + b; oa.w = accA[3] + b;
    ob.x = accB[0] + b; ob.y = accB[1] + b; ob.z = accB[2] + b; ob.w = accB[3] + b;
    float* dst = &attn[plane + (h0 + 2 * hq) * WWX + wq * 4];
    *reinterpret_cast<float4*>(dst)       = oa;
    *reinterpret_cast<float4*>(dst + WWX) = ob;
  }
}

// ---------------------------------------------------------------------------
// Kernel 2: pointwise 1x1 conv as WMMA f32 GEMM + bias + sigmoid + x*gate
// Block tile 128px x 128co, K-chunk 32; 8 waves = 4(M) x 2(N),
// each wave 2x4 tiles of v_wmma_f32_16x16x4_f32.
// A tile via global_load_async_to_lds_b128 (ASYNCcnt),
// B tile via Tensor Data Mover descriptor   (TENSORcnt).
// ---------------------------------------------------------------------------
__global__ __launch_bounds__(256) void pw_gemm_kernel(
    const float* __restrict__ attn, const float* __restrict__ pw_w,
    const float* __restrict__ pw_b, const float* __restrict__ x,
    float* __restrict__ out)
{
  __shared__ float smem[8448];      // sA 32x132 | sB 128x33 ; reused as ep 64x132
  float* sA = smem;                 // [ci][px], stride 132
  float* sB = smem + 32 * 132;      // [co][ci], stride 33

  const int tid    = threadIdx.x;
  const int lane   = tid & 31;
  const int wid    = tid >> 5;          // wave32
  const int wave_m = wid & 3;           // px offset 32*wave_m
  const int wave_n = wid >> 2;          // co offset 64*wave_n
  const int mlane  = lane & 15;
  const int khalf  = (lane >> 4) << 1;  // lanes 0-15 -> K 0/1, 16-31 -> K 2/3

  const int p0 = blockIdx.y * 128;      // pixel base
  const int c0 = blockIdx.x * 128;      // output-channel base

  v8f acc[2][4] = {};

  for (int kc = 0; kc < CCH; kc += 32) {
#if HAVE_TDM
    // ---- B tile via Tensor Data Mover: pw_w[c0..c0+128)[kc..kc+32) -------
    if (wid == 0) {
      unsigned long long gaddr =
          (unsigned long long)(const void*)(pw_w + (size_t)c0 * CCH + kc);
      unsigned int ldsOff = (unsigned int)(unsigned long long)AS3(sB);
      u32x4 g0;
      g0.x = 1u;                                   // count=1, user mode
      g0.y = ldsOff;                               // lds_addr (bytes)
      g0.z = (unsigned int)gaddr;                  // global_addr[31:0]
      g0.w = (unsigned int)((gaddr >> 32) & 0x01FFFFFFu) | (2u << 30); // type=2
      i32x8 g1;
      g1[0] = (int)((2u << 16) | (1u << 20) | (4u << 22));  // 4B, pad on, every 32 dw
      g1[1] = (int)(384u << 16);                   // tensor_dim0 = 384 (low 16)
      g1[2] = 0;                                   // tdim0 hi | tdim1 lo (1<<20 -> 0)
      g1[3] = (int)(16u | (32u << 16));            // tdim1 hi=16, tile_dim0=32
      g1[4] = 128;                                 // tile_dim1=128, tile_dim2=0
      g1[5] = 384;                                 // tensor_dim0_stride = 384
      g1[6] = 0;
      g1[7] = 0;
      i32x4 z4 = {0, 0, 0, 0};
#if defined(__clang_major__) && (__clang_major__ >= 23)
      i32x8 z8 = {0, 0, 0, 0, 0, 0, 0, 0};
      __builtin_amdgcn_tensor_load_to_lds(g0, g1, z4, z4, z8, 0);
#else
      __builtin_amdgcn_tensor_load_to_lds(g0, g1, z4, z4, 0);
#endif
    }
#else
    #pragma unroll
    for (int i = 0; i < 16; ++i) {
      int e   = tid + 256 * i;
      int col = e >> 5;
      int ci  = e & 31;
      sB[col * 33 + ci] = pw_w[(c0 + col) * CCH + kc + ci];
    }
#endif

#if HAVE_ASYNC_LDS
    // ---- A tile via async global->LDS b128 (handles image-boundary rows) --
    #pragma unroll
    for (int i = 0; i < 4; ++i) {
      int e4  = tid + 256 * i;            // 1024 float4 groups
      int ci  = e4 >> 5;
      int px4 = (e4 & 31) << 2;
      int p   = p0 + px4;
      int n   = p / HWP;                  // 4-aligned groups never cross images
      int hw  = p - n * HWP;
      const float* gp = attn + (n * CCH + kc + ci) * HWP + hw;
      __builtin_amdgcn_global_load_async_to_lds_b128(
          GAS1_V4(gp), LAS3_V4(sA + ci * 132 + px4), 0, 0);
    }
    WAIT_ASYNC0();
#else
    #pragma unroll
    for (int i = 0; i < 16; ++i) {
      int e   = tid + 256 * i;
      int ci  = e >> 7;
      int pxl = e & 127;
      int p   = p0 + pxl;
      int n   = p / HWP;
      int hw  = p - n * HWP;
      sA[ci * 132 + pxl] = attn[(n * CCH + kc + ci) * HWP + hw];
    }
#endif
#if HAVE_TDM
    if (wid == 0) { WAIT_TENSOR0(); }
#endif
    __syncthreads();

    #pragma unroll
    for (int k4 = 0; k4 < 32; k4 += 4) {
      v2f a[2], b[4];
      #pragma unroll
      for (int mt = 0; mt < 2; ++mt) {
        int px = wave_m * 32 + mt * 16 + mlane;
        a[mt].x = sA[(k4 + khalf + 0) * 132 + px];
        a[mt].y = sA[(k4 + khalf + 1) * 132 + px];
      }
      #pragma unroll
      for (int nt = 0; nt < 4; ++nt) {
        int co = wave_n * 64 + nt * 16 + mlane;
        b[nt].x = sB[co * 33 + k4 + khalf + 0];
        b[nt].y = sB[co * 33 + k4 + khalf + 1];
      }
      #pragma unroll
      for (int mt = 0; mt < 2; ++mt)
        #pragma unroll
        for (int nt = 0; nt < 4; ++nt)
          acc[mt][nt] = __builtin_amdgcn_wmma_f32_16x16x4_f32(
              false, a[mt], false, b[nt], (short)0, acc[mt][nt], false, false);
    }
    __syncthreads();
  }

  // Fused epilogue via LDS transpose (two co-halves of 64): coalesced x-loads
  // and out-stores along the pixel dimension.
  float* ep = smem;                      // [co 64][px 128], stride 132
  for (int half = 0; half < 2; ++half) {
    if (wave_n == half) {
      #pragma unroll
      for (int mt = 0; mt < 2; ++mt)
        #pragma unroll
        for (int nt = 0; nt < 4; ++nt)
          #pragma unroll
          for (int r = 0; r < 8; ++r) {
            int pxl = wave_m * 32 + mt * 16 + r + ((lane >> 4) << 3);
            int coh = nt * 16 + mlane;
            ep[coh * 132 + pxl] = acc[mt][nt][r];
          }
    }
    __syncthreads();
    #pragma unroll
    for (int g = 0; g < 8; ++g) {
      int idx = tid + 256 * g;           // 2048 float4 groups
      int coh = idx >> 5;                // 0..63
      int px4 = (idx & 31) * 4;
      int co  = c0 + half * 64 + coh;
      int p   = p0 + px4;
      int n   = p / HWP;
      int hw  = p - n * HWP;
      float bias = pw_b[co];
      float4 v  = *reinterpret_cast<float4*>(&ep[coh * 132 + px4]);
      int off   = (n * CCH + co) * HWP + hw;
      float4 xv = *reinterpret_cast<const float4*>(&x[off]);
      float4 o;
      o.x = xv.x * (1.0f / (1.0f + __expf(-(v.x + bias))));
      o.y = xv.y * (1.0f / (1.0f + __expf(-(v.y + bias))));
      o.z = xv.z * (1.0f / (1.0f + __expf(-(v.z + bias))));
      o.w = xv.w * (1.0f / (1.0f + __expf(-(v.w + bias))));
      *reinterpret_cast<float4*>(&out[off]) = o;
    }
    __syncthreads();
  }
}

// ---------------------------------------------------------------------------
extern "C" void kernel_launch(void* const* d_in, const int* in_sizes, int n_in,
                              void* d_out, int out_size, void* d_ws, size_t ws_size,
                              hipStream_t stream) {
  const float* x    = (const float*)d_in[0];
  const float* dw_w = (const float*)d_in[1];
  const float* dw_b = (const float*)d_in[2];
  const float* pw_w = (const float*)d_in[3];
  const float* pw_b = (const float*)d_in[4];
  float* out  = (float*)d_out;
  float* attn = (float*)d_ws;   // 32*384*3136 f32 = 154 MB scratch

  // depthwise 21x21 + bias -> attn
  dw_conv_kernel<<<dim3(NB * CCH, 2), 224, 0, stream>>>(x, dw_w, dw_b, attn);
  // pointwise GEMM (WMMA f32, TDM + async-LDS staging) + sigmoid gate -> out
  pw_gemm_kernel<<<dim3(3, 784), 256, 0, stream>>>(attn, pw_w, pw_b, x, out);
}